// NeumannSpectralBlock_19232863551756
// MI455X (gfx1250) — compile-verified
//
#include <hip/hip_runtime.h>
#include <math.h>

// ---------------------------------------------------------------------------
// FNO block for MI455X (gfx1250, wave32).
// Spectral path = truncated-DFT GEMMs (16 cols / 32 rows of modes retained),
// channel mixes + row-DFT on V_WMMA_F32_16X16X4_F32 (f32 reference precision;
// problem is HBM-bound: ~350MB traffic -> ~15us @ 23.3 TB/s, so keep VALU
// lean: all DFT twiddles come from a 128-entry LDS table since every angle
// is a multiple of 2*pi/128).
// Workspace requirement: ~104 MiB (27,262,976 floats).
// ---------------------------------------------------------------------------

typedef __attribute__((ext_vector_type(2))) float v2f;
typedef __attribute__((ext_vector_type(8))) float v8f;

#define BB  16
#define CC  64
#define HH  128
#define WW  128
#define M1M 16
#define KF  16          // retained frequency columns (rfft axis)
#define MR  32          // retained rows: 16 low + 16 high
#define HWN (HH * WW)

#define STEP (6.2831853071795864f / 128.0f)   // 2*pi/128

__device__ __forceinline__ v8f wmma_f32x4(v2f a, v2f b, v8f c) {
#if defined(__AMDGCN__) && __has_builtin(__builtin_amdgcn_wmma_f32_16x16x4_f32)
  // (neg_a, A, neg_b, B, c_mod, C, reuse_a, reuse_b)
  return __builtin_amdgcn_wmma_f32_16x16x4_f32(false, a, false, b, (short)0, c,
                                               false, false);
#else
  (void)a; (void)b;
  return c;   // host pass / builtin-missing fallback (never executed on device)
#endif
}

// 128-entry twiddle table: cs[i]=cos(2*pi*i/128), sn[i]=sin(2*pi*i/128).
__device__ __forceinline__ void fill_twiddle(float* cs, float* sn) {
  for (int i = threadIdx.x; i < 128; i += blockDim.x) {
    float s, c;
    __sincosf(STEP * (float)i, &s, &c);
    cs[i] = c; sn[i] = s;
  }
}

// ---------------------------------------------------------------------------
// 1x1 conv as GEMM: Out[b,o,hw] = sum_c Wm[o,c] * X[b,c,hw] + bias[o]
// One wave per 16(o) x 64(hw) super-tile: 4 accumulators share each weight
// A-fragment -> 64 WMMA/wave, B loads are immediate-offset clauses.
// grid: 2048 blocks x 256 thr = 16384 waves = 4 o-tiles * 4096 spatial tiles.
// ---------------------------------------------------------------------------
__global__ __launch_bounds__(256) void conv1x1_wmma(
    const float* __restrict__ X, const float* __restrict__ Wm,
    const float* __restrict__ bias, float* __restrict__ Out)
{
  const int wave = (blockIdx.x * blockDim.x + threadIdx.x) >> 5;
  const int lane = threadIdx.x & 31;
  const int lo = lane & 15, hi = lane >> 4;
  const int o0 = (wave & 3) * 16;
  const int stile = wave >> 2;                    // 0..4095
  const int n0 = (stile & ((HWN / 64) - 1)) * 64; // hw offset inside image
  const int b  = stile >> 8;                      // HWN/64 == 256
  const float* Xb = X + (size_t)b * CC * HWN + n0 + lo;

  v8f acc0 = {}, acc1 = {}, acc2 = {}, acc3 = {};
  for (int c0 = 0; c0 < CC; c0 += 4) {
    const int ca = c0 + 2 * hi;
    v2f a;
    a.x = Wm[(o0 + lo) * CC + ca];        // A[M=lo][K=2*hi + j] (shared by tiles)
    a.y = Wm[(o0 + lo) * CC + ca + 1];
    const float* p0 = Xb + (size_t)ca * HWN;        // row K=2*hi
    const float* p1 = Xb + (size_t)(ca + 1) * HWN;  // row K=2*hi+1
    v2f b0, b1, b2, b3;
    b0.x = p0[0];  b0.y = p1[0];
    b1.x = p0[16]; b1.y = p1[16];
    b2.x = p0[32]; b2.y = p1[32];
    b3.x = p0[48]; b3.y = p1[48];
    acc0 = wmma_f32x4(a, b0, acc0);
    acc1 = wmma_f32x4(a, b1, acc1);
    acc2 = wmma_f32x4(a, b2, acc2);
    acc3 = wmma_f32x4(a, b3, acc3);
  }
  float* Ob = Out + (size_t)b * CC * HWN + n0 + lo;
#pragma unroll
  for (int r = 0; r < 8; ++r) {
    const int o = o0 + r + 8 * hi;               // D row = r + 8*hi
    float* q = Ob + (size_t)o * HWN;
    const float bv = bias[o];
    q[0]  = acc0[r] + bv;
    q[16] = acc1[r] + bv;
    q[32] = acc2[r] + bv;
    q[48] = acc3[r] + bv;
  }
}

// ---------------------------------------------------------------------------
// Row DFT (truncated rfft over W, keep k=0..15):
//   F[row,k] = sum_w X[row,w] * exp(-2*pi*i*w*k/128)
// rows = B*C*H flattened; WMMA vs Fourier fragments from LDS twiddle table.
// grid: 1024 x 256 = 8192 waves = 8192 row-tiles of 16.
// ---------------------------------------------------------------------------
__global__ __launch_bounds__(256) void rowdft_wmma(
    const float* __restrict__ Xs, float* __restrict__ Fr, float* __restrict__ Fi)
{
  __shared__ float cs[128], sn[128];
  fill_twiddle(cs, sn);
  __syncthreads();

  const int wave = (blockIdx.x * blockDim.x + threadIdx.x) >> 5;
  const int lane = threadIdx.x & 31;
  const int lo = lane & 15, hi = lane >> 4;
  const int row0 = wave * 16;

  v8f ar = {}, ai = {};
  for (int w0 = 0; w0 < WW; w0 += 4) {
    const int wa = w0 + 2 * hi;
    v2f a, br, bi;
    a.x = Xs[(size_t)(row0 + lo) * WW + wa];
    a.y = Xs[(size_t)(row0 + lo) * WW + wa + 1];
    const int a0 = (wa * lo) & 127, a1 = ((wa + 1) * lo) & 127;
    br.x = cs[a0];  br.y = cs[a1];   // Re(e^{-i t}) =  cos t
    bi.x = -sn[a0]; bi.y = -sn[a1];  // Im(e^{-i t}) = -sin t
    ar = wmma_f32x4(a, br, ar);
    ai = wmma_f32x4(a, bi, ai);
  }
#pragma unroll
  for (int r = 0; r < 8; ++r) {
    const int row = row0 + r + 8 * hi;
    Fr[(size_t)row * KF + lo] = ar[r];
    Fi[(size_t)row * KF + lo] = ai[r];
  }
}

// ---------------------------------------------------------------------------
// Column DFT: XF[bc,mp,k] = sum_h F[bc,h,k] * exp(-2*pi*i*h*m/128),
// m = mp (mp<16) or mp+96 (rows -16..-1). One block per (b,c); F tile in LDS.
// ---------------------------------------------------------------------------
__global__ __launch_bounds__(512) void coldft(
    const float* __restrict__ Fr, const float* __restrict__ Fi,
    float* __restrict__ XFr, float* __restrict__ XFi)
{
  __shared__ float cs[128], sn[128];
  __shared__ float sr[HH * KF], si[HH * KF];
  fill_twiddle(cs, sn);
  const int bc = blockIdx.x;
  const float* pr = Fr + (size_t)bc * HH * KF;
  const float* pi = Fi + (size_t)bc * HH * KF;
  for (int i = threadIdx.x; i < HH * KF; i += blockDim.x) { sr[i] = pr[i]; si[i] = pi[i]; }
  __syncthreads();

  const int t = threadIdx.x;        // 512 = 32(mp) * 16(k)
  const int k = t & 15;
  const int mp = t >> 4;
  const int m = (mp < 16) ? mp : (mp + 96);
  float accr = 0.f, acci = 0.f;
  for (int h = 0; h < HH; ++h) {
    const int a = (h * m) & 127;
    const float c = cs[a], s = sn[a];
    const float xr = sr[h * KF + k], xi = si[h * KF + k];
    accr += xr * c + xi * s;        // Re(x * e^{-i t})
    acci += xi * c - xr * s;        // Im(x * e^{-i t})
  }
  XFr[((size_t)bc * MR + mp) * KF + k] = accr;
  XFi[((size_t)bc * MR + mp) * KF + k] = acci;
}

// ---------------------------------------------------------------------------
// Mode mixing: out[b,o,mp,k] = sum_i XF[b,i,mp,k] * W[i,o,mm,k]  (complex)
// Complex GEMM (M=B=16, N=o, K=i) via 4 real WMMA accumulators.
// grid: 256 x 256 = 2048 waves = 512 modes * 4 o-tiles.
// ---------------------------------------------------------------------------
__global__ __launch_bounds__(256) void modemix_wmma(
    const float* __restrict__ XFr, const float* __restrict__ XFi,
    const float* __restrict__ w1r, const float* __restrict__ w1i,
    const float* __restrict__ w2r, const float* __restrict__ w2i,
    float* __restrict__ Gr, float* __restrict__ Gi)
{
  const int wave = (blockIdx.x * blockDim.x + threadIdx.x) >> 5;
  const int lane = threadIdx.x & 31;
  const int lo = lane & 15, hi = lane >> 4;
  const int o0 = (wave & 3) * 16;
  const int mode = wave >> 2;       // 0..511
  const int k = mode & 15;
  const int mp = mode >> 4;         // 0..31
  const float* Wr; const float* Wi; int mm;
  if (mp < M1M) { Wr = w1r; Wi = w1i; mm = mp; }
  else          { Wr = w2r; Wi = w2i; mm = mp - M1M; }

  v8f drr = {}, dii = {}, dri = {}, dir_ = {};
  for (int c0 = 0; c0 < CC; c0 += 4) {
    const int ca = c0 + 2 * hi;
    const size_t xb0 = (((size_t)lo * CC + ca) * MR + mp) * KF + k;  // b=lo
    const size_t xst = (size_t)MR * KF;                               // +1 channel
    v2f arr, aii, brr, bii;
    arr.x = XFr[xb0]; arr.y = XFr[xb0 + xst];
    aii.x = XFi[xb0]; aii.y = XFi[xb0 + xst];
    const size_t wb0 = (((size_t)ca * CC + (o0 + lo)) * M1M + mm) * KF + k;
    const size_t wst = (size_t)CC * M1M * KF;                         // +1 channel
    brr.x = Wr[wb0]; brr.y = Wr[wb0 + wst];
    bii.x = Wi[wb0]; bii.y = Wi[wb0 + wst];
    drr  = wmma_f32x4(arr, brr, drr);
    dii  = wmma_f32x4(aii, bii, dii);
    dri  = wmma_f32x4(arr, bii, dri);
    dir_ = wmma_f32x4(aii, brr, dir_);
  }
#pragma unroll
  for (int r = 0; r < 8; ++r) {
    const int b = r + 8 * hi;
    const size_t oidx = (((size_t)b * CC + (o0 + lo)) * MR + mp) * KF + k;
    Gr[oidx] = drr[r] - dii[r];   // Re: Ar*Br - Ai*Bi
    Gi[oidx] = dri[r] + dir_[r];  // Im: Ar*Bi + Ai*Br
  }
}

// ---------------------------------------------------------------------------
// Inverse column DFT: Y[bo,h,k] = sum_mp G[bo,mp,k] * exp(+2*pi*i*h*m/128)
// One block per (b,o); G tile (32x16 complex) + twiddles in LDS.
// ---------------------------------------------------------------------------
__global__ __launch_bounds__(256) void invcol(
    const float* __restrict__ Gr, const float* __restrict__ Gi,
    float* __restrict__ Yr, float* __restrict__ Yi)
{
  __shared__ float cs[128], sn[128];
  __shared__ float gr[MR * KF], gi[MR * KF];
  fill_twiddle(cs, sn);
  const int bo = blockIdx.x;
  for (int i = threadIdx.x; i < MR * KF; i += blockDim.x) {
    gr[i] = Gr[(size_t)bo * MR * KF + i];
    gi[i] = Gi[(size_t)bo * MR * KF + i];
  }
  __syncthreads();
  for (int t = threadIdx.x; t < HH * KF; t += blockDim.x) {
    const int k = t & 15;
    const int h = t >> 4;
    float accr = 0.f, acci = 0.f;
    for (int mp = 0; mp < MR; ++mp) {
      const int m = (mp < 16) ? mp : (mp + 96);
      const int a = (h * m) & 127;
      const float c = cs[a], s = sn[a];          // e^{+i t}
      const float xr = gr[mp * KF + k], xi = gi[mp * KF + k];
      accr += xr * c - xi * s;
      acci += xr * s + xi * c;
    }
    Yr[(size_t)bo * HH * KF + t] = accr;
    Yi[(size_t)bo * HH * KF + t] = acci;
  }
}

// ---------------------------------------------------------------------------
// Inverse row rDFT (Hermitian: c0=1, ck=2 for 1<=k<=15; no Nyquist term)
// + 1/(H*W) norm + skip + exact-erf GELU. One block per (b,o) image with the
// 128x16 complex spectrum staged in LDS (read once instead of per-row).
// ---------------------------------------------------------------------------
__global__ __launch_bounds__(256) void fuse_out(
    const float* __restrict__ Skip, const float* __restrict__ Yr,
    const float* __restrict__ Yi, float* __restrict__ Out)
{
  __shared__ float cs[128], sn[128];
  __shared__ float yr[HH * KF], yi[HH * KF];
  fill_twiddle(cs, sn);
  const int bo = blockIdx.x;                    // b*CC + o
  for (int i = threadIdx.x; i < HH * KF; i += blockDim.x) {
    yr[i] = Yr[(size_t)bo * HH * KF + i];
    yi[i] = Yi[(size_t)bo * HH * KF + i];
  }
  __syncthreads();

  const float* skp = Skip + (size_t)bo * HWN;
  float* outp = Out + (size_t)bo * HWN;
  for (int idx = threadIdx.x; idx < HWN; idx += blockDim.x) {
    const int h = idx >> 7;
    const int w = idx & 127;
    float acc = yr[h * KF];                     // k=0: cos=1, sin=0 (Re only)
#pragma unroll
    for (int k = 1; k < KF; ++k) {
      const int a = (w * k) & 127;
      acc += 2.f * (yr[h * KF + k] * cs[a] - yi[h * KF + k] * sn[a]);
    }
    const float x = skp[idx] + acc * (1.f / (float)(HH * WW));
    outp[idx] = 0.5f * x * (1.f + erff(x * 0.70710678f));
  }
}

// ---------------------------------------------------------------------------
extern "C" void kernel_launch(void* const* d_in, const int* in_sizes, int n_in,
                              void* d_out, int out_size, void* d_ws, size_t ws_size,
                              hipStream_t stream) {
  (void)in_sizes; (void)n_in; (void)out_size; (void)ws_size;

  const float* h_in   = (const float*)d_in[0];
  const float* u_inc  = (const float*)d_in[1];
  const float* w_loc  = (const float*)d_in[2];
  const float* b_loc  = (const float*)d_in[3];
  const float* w_skip = (const float*)d_in[4];
  const float* b_skip = (const float*)d_in[5];
  const float* w1r    = (const float*)d_in[6];
  const float* w1i    = (const float*)d_in[7];
  const float* w2r    = (const float*)d_in[8];
  const float* w2i    = (const float*)d_in[9];
  float* out = (float*)d_out;
  float* ws  = (float*)d_ws;

  // Workspace layout (floats). Total 27,262,976 floats (~104 MiB).
  float* scat = ws;                              // B*C*H*W  (also reused for skip)
  float* Fr   = scat + (size_t)BB * CC * HWN;    // B*C*H*16
  float* Fi_  = Fr   + (size_t)BB * CC * HH * KF;
  float* XFr  = Fi_  + (size_t)BB * CC * HH * KF; // B*C*32*16
  float* XFi  = XFr  + (size_t)BB * CC * MR * KF;
  float* Gr   = XFi  + (size_t)BB * CC * MR * KF;
  float* Gi   = Gr   + (size_t)BB * CC * MR * KF;
  float* Yr   = Gi   + (size_t)BB * CC * MR * KF; // B*C*H*16
  float* Yi   = Yr   + (size_t)BB * CC * HH * KF;

  conv1x1_wmma<<<2048, 256, 0, stream>>>(h_in, w_loc, b_loc, scat);
  rowdft_wmma <<<1024, 256, 0, stream>>>(scat, Fr, Fi_);
  coldft      <<<BB * CC, 512, 0, stream>>>(Fr, Fi_, XFr, XFi);
  modemix_wmma<<<256, 256, 0, stream>>>(XFr, XFi, w1r, w1i, w2r, w2i, Gr, Gi);
  invcol      <<<BB * CC, 256, 0, stream>>>(Gr, Gi, Yr, Yi);
  // scattered buffer is dead after rowdft -> reuse it for the skip GEMM.
  conv1x1_wmma<<<2048, 256, 0, stream>>>(u_inc, w_skip, b_skip, scat);
  fuse_out    <<<BB * CC, 256, 0, stream>>>(scat, Yr, Yi, out);
}